// UltraEfficientSpikeNeuron_43860206027117
// MI455X (gfx1250) — compile-verified
//
#include <hip/hip_runtime.h>
#include <hip/hip_fp16.h>

// LIF spike-neuron scan, MI455X (gfx1250).
//   x:  [B=32, S=1024, N=2048] fp32
//   m0: [B, N] fp32
// out: spikes [B,S,N] fp16  ++  m_final [B,N] fp32   (flat-concatenated in d_out)
//
// Memory-bound: ~402 MB @ 23.3 TB/s => ~17 us floor. Only B*N = 65536 serial
// chains exist, so MLP comes from a 32-deep async global->LDS pipeline
// (ASYNCcnt / s_wait_asynccnt). Steady-state loop fully unrolled by STAGES so
// LDS reads use ds immediate offsets and the async LDS-address VGPR is
// loop-invariant (INST_OFFSET adds to both LDS and global addresses).

#define B_ 32
#define S_ 1024
#define N_ 2048

#define TPB    128      // threads per block (4 wave32s)
#define VEC    2        // columns per thread -> b64 async loads
#define STAGES 32       // pipeline depth (ASYNCcnt max 63; wait imm 31)

#define STAGE_BYTES   (TPB * VEC * 4)     // 1024 B of LDS per stage
#define ROW_BYTES     (N_ * 4)            // 8192 B per timestep row slice stride

__global__ __launch_bounds__(TPB)
void lif_scan_kernel(const float* __restrict__ x,
                     const float* __restrict__ m_init,
                     unsigned int* __restrict__ spikes_u32,  // packed fp16 pairs
                     float* __restrict__ m_out)
{
    // 32 stages * 128 lanes * 8B = 32 KB LDS
    __shared__ __align__(16) float buf[STAGES * TPB * VEC];

    const int tid          = threadIdx.x;
    const int blocks_per_b = N_ / (TPB * VEC);                 // 8
    const int b            = blockIdx.x / blocks_per_b;
    const int n0           = (blockIdx.x % blocks_per_b) * (TPB * VEC);
    const int nn           = n0 + tid * VEC;                   // first of my 2 cols

    // Block-uniform global base (GVS mode) + per-lane 32-bit byte offset.
    const float* xbase = x + (size_t)b * S_ * N_ + n0;
    const unsigned lane_off = (unsigned)(tid * VEC * 4);

    // Wave-relative LDS byte address (low 32 bits of the generic pointer).
    const unsigned lane_lds =
        (unsigned)(size_t)(void*)&buf[0] + (unsigned)(tid * VEC * 4);

    // ---------------- prologue: fill the async pipeline ----------------
    // INST_OFFSET (= u*STAGE_BYTES) is added to BOTH the LDS and the global
    // address, so the LDS address VGPR stays fixed at lane_lds and the global
    // VGPR offset is compensated by -u*STAGE_BYTES.
    #pragma unroll
    for (int u = 0; u < STAGES; ++u) {
        unsigned vo = lane_off + (unsigned)(u * ROW_BYTES - u * STAGE_BYTES);
        asm volatile("global_load_async_to_lds_b64 %0, %1, %2 offset:%3 th:TH_LOAD_NT"
                     :: "v"(lane_lds), "v"(vo), "s"(xbase),
                        "i"(u * STAGE_BYTES)
                     : "memory");
    }

    // membrane state in VGPRs
    float m0 = m_init[(size_t)b * N_ + nn];
    float m1 = m_init[(size_t)b * N_ + nn + 1];

    // packed-fp16 spike output (uint = 2 halves); nn even -> 4B aligned
    unsigned int* sp = spikes_u32 + ((size_t)b * S_ * N_ + (size_t)nn) / 2;

    // ---------------- steady state: 31 outer x 32 unrolled ----------------
    for (int outer = 0; outer < (S_ - STAGES) / STAGES; ++outer) {
        #pragma unroll
        for (int u = 0; u < STAGES; ++u) {
            const int t = outer * STAGES + u;          // stage index == u

            // oldest outstanding stage (t % STAGES == u) has landed in LDS
            asm volatile("s_wait_asynccnt 31" ::: "memory");

            const int base = u * (TPB * VEC) + tid * VEC;  // imm-folded ds offset
            float x0 = buf[base + 0];
            float x1 = buf[base + 1];

            m0 = fmaf(0.8f, m0, x0);
            m1 = fmaf(0.8f, m1, x1);
            const bool s0 = m0 > 0.5f;
            const bool s1 = m1 > 0.5f;
            // fp16 1.0 == 0x3C00 : build packed pair bitwise, no cvt needed
            unsigned pack = (s0 ? 0x00003C00u : 0u) | (s1 ? 0x3C000000u : 0u);
            __builtin_nontemporal_store(pack, &sp[(size_t)t * (N_ / 2)]);
            m0 = s0 ? 0.0f : m0;
            m1 = s1 ? 0.0f : m1;

            // refill stage u with timestep t + STAGES
            unsigned vo = lane_off +
                (unsigned)((t + STAGES) * ROW_BYTES - u * STAGE_BYTES);
            asm volatile("global_load_async_to_lds_b64 %0, %1, %2 offset:%3 th:TH_LOAD_NT"
                         :: "v"(lane_lds), "v"(vo), "s"(xbase),
                            "i"(u * STAGE_BYTES)
                         : "memory");
        }
    }

    // ---------------- drain: last 32 steps, all data already requested ------
    asm volatile("s_wait_asynccnt 0" ::: "memory");
    #pragma unroll
    for (int u = 0; u < STAGES; ++u) {
        const int t = (S_ - STAGES) + u;               // stage index == u
        const int base = u * (TPB * VEC) + tid * VEC;
        float x0 = buf[base + 0];
        float x1 = buf[base + 1];

        m0 = fmaf(0.8f, m0, x0);
        m1 = fmaf(0.8f, m1, x1);
        const bool s0 = m0 > 0.5f;
        const bool s1 = m1 > 0.5f;
        unsigned pack = (s0 ? 0x00003C00u : 0u) | (s1 ? 0x3C000000u : 0u);
        __builtin_nontemporal_store(pack, &sp[(size_t)t * (N_ / 2)]);
        m0 = s0 ? 0.0f : m0;
        m1 = s1 ? 0.0f : m1;
    }

    // final membrane potential (fp32), b64 store
    float2 mf = make_float2(m0, m1);
    *(float2*)&m_out[(size_t)b * N_ + nn] = mf;
}

extern "C" void kernel_launch(void* const* d_in, const int* in_sizes, int n_in,
                              void* d_out, int out_size, void* d_ws, size_t ws_size,
                              hipStream_t stream) {
    (void)in_sizes; (void)n_in; (void)out_size; (void)d_ws; (void)ws_size;

    const float* x  = (const float*)d_in[0];
    const float* m0 = (const float*)d_in[1];

    // output 0: spikes [B,S,N] fp16; output 1: m_final [B,N] fp32 (flat concat)
    unsigned int* spikes_u32 = (unsigned int*)d_out;
    float* m_out = (float*)((char*)d_out + (size_t)B_ * S_ * N_ * sizeof(__half));

    dim3 grid(B_ * (N_ / (TPB * VEC)));   // 256 blocks
    dim3 block(TPB);                      // 128 threads = 4 waves
    lif_scan_kernel<<<grid, block, 0, stream>>>(x, m0, spikes_u32, m_out);
}